// MultiBoxLoss_50603304681691
// MI455X (gfx1250) — compile-verified
//
#include <hip/hip_runtime.h>
#include <float.h>

// Problem constants (match reference)
#define BB 32
#define PP 24576
#define OO 16
#define CC 81

typedef __attribute__((ext_vector_type(2))) float v2f;
typedef __attribute__((ext_vector_type(8))) float v8f;

#if defined(__has_builtin)
#if __has_builtin(__builtin_amdgcn_global_load_async_to_lds_b128)
#define HAVE_ASYNC_LDS 1
#endif
#endif

#ifdef HAVE_ASYNC_LDS
// Exact parameter types per hipcc diagnostic:
//   arg0: __attribute__((vector_size(16))) int __device__ *   (global, AS1)
//   arg1: same vector type in __shared__ (AS3)
typedef int v4i_gcc __attribute__((vector_size(16)));
typedef __attribute__((address_space(1))) v4i_gcc g_v4i;
typedef __attribute__((address_space(3))) v4i_gcc l_v4i;
#endif

// ---------------------------------------------------------------------------
// Wave-level exact-f32 sum using V_WMMA_F32_16X16X4_F32.
// A = 16x4 (lane m holds K=0,1 ; lane m+16 holds K=2,3), B = ones(4x16).
// D[m][n] = v[m] + v[m+16]  (replicated across n).
// Sum of the 8 accumulator VGPRs = rows 0..7 (lanes<16) / rows 8..15
// (lanes>=16); one xor-16 shuffle gives the full 32-lane sum on every lane.
// Requires full EXEC (all call sites are convergent).
// ---------------------------------------------------------------------------
__device__ __forceinline__ float wave_sum_wmma(float v) {
  v2f a; a.x = v;   a.y = 0.0f;
  v2f b; b.x = 1.0f; b.y = 1.0f;
  v8f c = {};
  c = __builtin_amdgcn_wmma_f32_16x16x4_f32(false, a, false, b, (short)0, c,
                                            false, false);
  float s = c[0] + c[1] + c[2] + c[3] + c[4] + c[5] + c[6] + c[7];
  s += __shfl_xor(s, 16, 32);
  return s;
}

// Block-wide sum (result broadcast to all threads). s_red must be float[32].
__device__ __forceinline__ float block_sum(float v, float* s_red) {
  float w  = wave_sum_wmma(v);
  int lane = threadIdx.x & 31;
  int wid  = threadIdx.x >> 5;
  int nw   = (blockDim.x + 31) >> 5;
  if (lane == 0) s_red[wid] = w;
  __syncthreads();
  if (wid == 0) {                      // wave-uniform branch: EXEC stays full
    float x = (lane < nw) ? s_red[lane] : 0.0f;
    x = wave_sum_wmma(x);
    if (lane == 0) s_red[0] = x;
  }
  __syncthreads();
  float t = s_red[0];
  __syncthreads();
  return t;
}

__device__ __forceinline__ float smooth_l1(float d) {
  float ad = fabsf(d);
  return (ad < 1.0f) ? 0.5f * d * d : ad - 0.5f;
}

// ---------------------------------------------------------------------------
// Kernel 0: zero the accumulators.
// acc[0]=loc_sum, acc[1]=pos_ce_sum, acc[2]=hardneg_sum, npt = total positives
// ---------------------------------------------------------------------------
__global__ void init_kernel(float* acc, int* npt) {
  if (threadIdx.x < 3) acc[threadIdx.x] = 0.0f;
  if (threadIdx.x == 3) *npt = 0;
}

// ---------------------------------------------------------------------------
// Kernel 1: prior<->GT matching. One block per image, assignment kept in LDS.
// boxes: [B,O,4] corner xyxy, priors: [P,4] cxcywh.
// assign[b,p] = object index (0..15) or -1 for background.
// ---------------------------------------------------------------------------
__global__ void match_kernel(const float* __restrict__ boxes,
                             const float* __restrict__ priors,
                             signed char* __restrict__ assign,
                             int* __restrict__ npos,
                             int* __restrict__ npos_tot) {
  __shared__ float4 s_box[OO];
  __shared__ float  s_area[OO];
  __shared__ unsigned long long s_best[OO];   // (iou_bits<<32)|prior
  __shared__ signed char s_as[PP];            // 24 KB
  __shared__ float  s_red[32];

  const int b = blockIdx.x, tid = threadIdx.x;
  if (tid < OO) {
    float4 bx   = ((const float4*)boxes)[b * OO + tid];
    s_box[tid]  = bx;
    s_area[tid] = fmaxf(bx.z - bx.x, 0.0f) * fmaxf(bx.w - bx.y, 0.0f);
    s_best[tid] = 0ull;
  }
  __syncthreads();

  for (int p = tid; p < PP; p += blockDim.x) {
    float4 pr = ((const float4*)priors)[p];
    float px1 = pr.x - 0.5f * pr.z, py1 = pr.y - 0.5f * pr.w;
    float px2 = pr.x + 0.5f * pr.z, py2 = pr.y + 0.5f * pr.w;
    float pa  = pr.z * pr.w;
    float best = 0.0f; int bo = -1;
#pragma unroll
    for (int o = 0; o < OO; ++o) {
      float4 bx = s_box[o];
      float iw = fminf(px2, bx.z) - fmaxf(px1, bx.x);
      float ih = fminf(py2, bx.w) - fmaxf(py1, bx.y);
      float inter = fmaxf(iw, 0.0f) * fmaxf(ih, 0.0f);
      float iou = inter / (pa + s_area[o] - inter);
      if (iou > best) { best = iou; bo = o; }
      unsigned long long key =
          ((unsigned long long)__float_as_uint(iou) << 32) | (unsigned)p;
      atomicMax(&s_best[o], key);
    }
    s_as[p] = (best >= 0.5f) ? (signed char)bo : (signed char)-1;
  }
  __syncthreads();

  // Force-match: best prior per object, sequential for determinism.
  if (tid == 0) {
    for (int o = 0; o < OO; ++o) {
      int pbest = (int)(s_best[o] & 0xffffffffull);
      s_as[pbest] = (signed char)o;
    }
  }
  __syncthreads();

  // Write out (packed int stores) and count positives from the same words:
  // byte == 0xFF (-1) is background; bytes 0..15 (MSB clear) are positives.
  int* out4 = (int*)(assign + (size_t)b * PP);
  const int* in4 = (const int*)s_as;
  float cnt = 0.0f;
  for (int i = tid; i < PP / 4; i += blockDim.x) {
    int wrd = in4[i];
    out4[i] = wrd;
    unsigned posmask = ~((unsigned)wrd) & 0x80808080u;
    cnt += (float)__popc(posmask);
  }
  float tot = block_sum(cnt, s_red);
  if (tid == 0) {
    int c = (int)(tot + 0.5f);
    npos[b] = c;
    atomicAdd(npos_tot, c);
  }
}

// ---------------------------------------------------------------------------
// Kernel 2: localization loss (smooth-L1 on positives, gcxgcy encoding).
// One thread per prior; grid covers B*P exactly.
// ---------------------------------------------------------------------------
__global__ void loc_loss_kernel(const float* __restrict__ locs,
                                const float* __restrict__ boxes,
                                const float* __restrict__ priors,
                                const signed char* __restrict__ assign,
                                float* __restrict__ acc) {
  __shared__ float s_red[32];
  int g = blockIdx.x * blockDim.x + threadIdx.x;   // [0, B*P)
  int b = g / PP, p = g - b * PP;
  float l = 0.0f;
  int a = assign[g];
  if (a >= 0) {
    float4 bx = ((const float4*)boxes)[b * OO + a];
    float4 pr = ((const float4*)priors)[p];
    float cx = 0.5f * (bx.x + bx.z), cy = 0.5f * (bx.y + bx.w);
    float w  = bx.z - bx.x,          h  = bx.w - bx.y;
    float4 pl = ((const float4*)locs)[g];
    float d0 = pl.x - (cx - pr.x) / (pr.z * 0.1f);
    float d1 = pl.y - (cy - pr.y) / (pr.w * 0.1f);
    float d2 = pl.z - 5.0f * logf(w / pr.z);
    float d3 = pl.w - 5.0f * logf(h / pr.w);
    l = smooth_l1(d0) + smooth_l1(d1) + smooth_l1(d2) + smooth_l1(d3);
  }
  float tot = block_sum(l, s_red);
  if (threadIdx.x == 0) atomicAdd(acc + 0, tot);
}

// ---------------------------------------------------------------------------
// Kernel 3: cross-entropy per prior. One wave per prior: 32 lanes read the 81
// contiguous class scores coalesced, shuffle-reduce max / sum-exp / label pick.
// This is the HBM-bound kernel (255 MB of scores, read once).
// Positives: accumulate CE into acc[1], store 0 in ce_arr (so they never rank
// as hard negatives). Negatives: store CE in ce_arr.
// ---------------------------------------------------------------------------
__global__ void conf_kernel(const float* __restrict__ scores,
                            const int* __restrict__ labels,
                            const signed char* __restrict__ assign,
                            float* __restrict__ ce_arr,
                            float* __restrict__ acc) {
  __shared__ float s_red[32];
  const int lane = threadIdx.x & 31;
  const int wv = (blockIdx.x * blockDim.x + threadIdx.x) >> 5;  // prior id
  const int b = wv / PP;
  const float* s = scores + (size_t)wv * CC;

  float v0 = s[lane];
  float v1 = s[lane + 32];
  float v2 = (lane < CC - 64) ? s[lane + 64] : -FLT_MAX;

  float m = fmaxf(fmaxf(v0, v1), v2);
#pragma unroll
  for (int off = 16; off > 0; off >>= 1) m = fmaxf(m, __shfl_xor(m, off, 32));

  float se = expf(v0 - m) + expf(v1 - m) +
             ((lane < CC - 64) ? expf(v2 - m) : 0.0f);
#pragma unroll
  for (int off = 16; off > 0; off >>= 1) se += __shfl_xor(se, off, 32);

  int a = assign[wv];
  int lab = (a >= 0) ? labels[b * OO + a] : 0;
  float sl = 0.0f;
  if (lane == lab) sl = v0;
  if (lane + 32 == lab) sl = v1;
  if (lane + 64 == lab && lane < CC - 64) sl = v2;
#pragma unroll
  for (int off = 16; off > 0; off >>= 1) sl += __shfl_xor(sl, off, 32);

  float ce = m + logf(se) - sl;
  bool pos = (a >= 0);
  if (lane == 0) ce_arr[wv] = pos ? 0.0f : ce;

  float contrib = (lane == 0 && pos) ? ce : 0.0f;
  float tot = block_sum(contrib, s_red);
  if (threadIdx.x == 0) atomicAdd(acc + 1, tot);
}

// ---------------------------------------------------------------------------
// Kernel 4: hard-negative mining. One block per image; the image's 24576 CE
// values (96 KB) live entirely in LDS (CDNA5: 320 KB/WGP), staged via the
// async global->LDS DMA path (ASYNCcnt) when available. Top-k sum found by
// 30-step threshold bisection over LDS, then sum of values above threshold.
// ---------------------------------------------------------------------------
__global__ void hardneg_kernel(const float* __restrict__ ce_arr,
                               const int* __restrict__ npos,
                               float* __restrict__ acc) {
  extern __shared__ float s_ce[];           // PP floats (dynamic LDS)
  __shared__ float s_red[32];
  __shared__ float s_bcast;
  const int b = blockIdx.x, tid = threadIdx.x, nt = blockDim.x;
  const float* src = ce_arr + (size_t)b * PP;

  float mx = 0.0f;
#ifdef HAVE_ASYNC_LDS
  // 16B-aligned: ce buffer offset & per-image stride are multiples of 16,
  // every thread issues exactly PP/(4*nt) b128 ops (uniform, full EXEC).
  for (int i = tid * 4; i < PP; i += nt * 4) {
    __builtin_amdgcn_global_load_async_to_lds_b128(
        (g_v4i*)(src + i), (l_v4i*)(s_ce + i), 0, 0);
  }
#if __has_builtin(__builtin_amdgcn_s_wait_asynccnt)
  __builtin_amdgcn_s_wait_asynccnt(0);
#else
  asm volatile("s_wait_asynccnt 0x0" ::: "memory");
#endif
  __syncthreads();
  for (int i = tid; i < PP; i += nt) mx = fmaxf(mx, s_ce[i]);
#else
  for (int i = tid; i < PP; i += nt) {
    float v = src[i];
    s_ce[i] = v;
    mx = fmaxf(mx, v);
  }
#endif

#pragma unroll
  for (int off = 16; off > 0; off >>= 1) mx = fmaxf(mx, __shfl_xor(mx, off, 32));
  int lane = tid & 31, wid = tid >> 5, nw = nt >> 5;
  if (lane == 0) s_red[wid] = mx;
  __syncthreads();
  if (wid == 0) {
    float x = (lane < nw) ? s_red[lane] : 0.0f;
#pragma unroll
    for (int off = 16; off > 0; off >>= 1) x = fmaxf(x, __shfl_xor(x, off, 32));
    if (lane == 0) s_bcast = x;
  }
  __syncthreads();
  float hi = s_bcast, lo = 0.0f;
  const float k = (float)(3 * npos[b]);     // NEG_POS_RATIO = 3
  __syncthreads();

  for (int it = 0; it < 30; ++it) {
    float mid = 0.5f * (lo + hi);
    float c = 0.0f;
    for (int i = tid; i < PP; i += nt) c += (s_ce[i] > mid) ? 1.0f : 0.0f;
    float cnt = block_sum(c, s_red);        // broadcast -> uniform update
    if (cnt > k) lo = mid; else hi = mid;
  }

  float ssum = 0.0f, c2 = 0.0f;
  for (int i = tid; i < PP; i += nt) {
    float v = s_ce[i];
    if (v > hi) { ssum += v; c2 += 1.0f; }
  }
  float total = block_sum(ssum, s_red);
  float cnt2  = block_sum(c2, s_red);
  if (tid == 0) {
    float rem = k - cnt2;
    if (rem < 0.0f) rem = 0.0f;
    atomicAdd(acc + 2, total + rem * hi);   // ties at threshold ~= hi
  }
}

// ---------------------------------------------------------------------------
// Kernel 5: final scalar loss = conf + loc.
// conf = (pos_ce + hard_neg) / N_pos ; loc = smoothL1_sum / (4*N_pos) (mean).
// ---------------------------------------------------------------------------
__global__ void final_kernel(const float* __restrict__ acc,
                             const int* __restrict__ npt,
                             float* __restrict__ out, int n) {
  int np = *npt; if (np < 1) np = 1;
  float N = (float)np;
  float loss = (acc[1] + acc[2]) / N + acc[0] / (4.0f * N);
  for (int i = threadIdx.x; i < n; i += blockDim.x) out[i] = loss;
}

// ---------------------------------------------------------------------------
extern "C" void kernel_launch(void* const* d_in, const int* in_sizes, int n_in,
                              void* d_out, int out_size, void* d_ws,
                              size_t ws_size, hipStream_t stream) {
  const float* locs = nullptr;
  const float* scores = nullptr;
  const float* boxes = nullptr;
  const int* labels = nullptr;
  const float* priors = nullptr;
  for (int i = 0; i < n_in; ++i) {
    int s = in_sizes[i];
    if (s == BB * PP * 4)       locs   = (const float*)d_in[i];
    else if (s == BB * PP * CC) scores = (const float*)d_in[i];
    else if (s == BB * OO * 4)  boxes  = (const float*)d_in[i];
    else if (s == BB * OO)      labels = (const int*)d_in[i];
    else if (s == PP * 4)       priors = (const float*)d_in[i];
  }

  // Workspace layout (~3.9 MB)
  float* acc = (float*)d_ws;                         // [3]
  int* npt   = (int*)d_ws + 3;                       // total positives
  int* npos  = (int*)d_ws + 4;                       // [B]
  signed char* assign = (signed char*)(npos + BB);   // [B*P]
  float* ce = (float*)(assign + (size_t)BB * PP);    // [B*P], 16B-aligned

  init_kernel<<<1, 64, 0, stream>>>(acc, npt);
  match_kernel<<<BB, 256, 0, stream>>>(boxes, priors, assign, npos, npt);
  loc_loss_kernel<<<BB * PP / 256, 256, 0, stream>>>(locs, boxes, priors,
                                                     assign, acc);
  conf_kernel<<<BB * PP / 8, 256, 0, stream>>>(scores, labels, assign, ce, acc);
  hardneg_kernel<<<BB, 1024, PP * sizeof(float), stream>>>(ce, npos, acc);
  final_kernel<<<1, 256, 0, stream>>>(acc, npt, (float*)d_out, out_size);
}